// ProbAttention_7799660609832
// MI455X (gfx1250) — compile-verified
//
#include <hip/hip_runtime.h>
#include <math.h>

// ---------------- CDNA5 types ----------------
typedef __attribute__((ext_vector_type(2))) float v2f;
typedef __attribute__((ext_vector_type(8))) float v8f;

#define HAS_WMMA_F32 __has_builtin(__builtin_amdgcn_wmma_f32_16x16x4_f32)

static __device__ inline v8f wmma_f32_16x16x4(v2f a, v2f b, v8f c) {
#if HAS_WMMA_F32
  // (neg_a, A, neg_b, B, c_mod, C, reuse_a, reuse_b) -- exact f32 MAC, D = A(16x4) * B(4x16) + C
  return __builtin_amdgcn_wmma_f32_16x16x4_f32(false, a, false, b, (short)0, c, false, false);
#else
  // placeholder so the file still compiles; histogram (wmma==0) will reveal this path
  c[0] += a.x * b.x;
  return c;
#endif
}

// ---------------- problem constants ----------------
constexpr int B = 4, L = 2048, H = 8, D = 64, S = 2048;
constexpr int SAMPK = 38;           // int(5*ln(2048))
constexpr int NTOP = 38;
constexpr int NTOP_PAD = 48;        // 3 tiles of 16 rows
constexpr int CHUNK = 128, NCHUNK = L / CHUNK;
constexpr float SCALE = 0.125f;     // 1/sqrt(64)

// =====================================================================
// Kernel 1: sparsity metric  M[b,h,l] = max_j(q.k_sample) - sum_j/S
// (index sample: deterministic hash, shared across b,h like the reference)
// =====================================================================
__global__ __launch_bounds__(256) void k_sample_m(const float* __restrict__ Q,
                                                  const float* __restrict__ K,
                                                  float* __restrict__ M) {
  int gid = blockIdx.x * 256 + threadIdx.x;  // B*H*L threads
  int l = gid & (L - 1);
  int bh = gid / L;
  int h = bh & (H - 1);
  int b = bh / H;
  const float* qrow = Q + ((size_t)((b * L + l) * H + h)) * D;
  float4 q[16];
#pragma unroll
  for (int t = 0; t < 16; ++t) q[t] = ((const float4*)qrow)[t];
  float mx = -__builtin_inff();
  float sm = 0.f;
  for (int j = 0; j < SAMPK; ++j) {
    unsigned x = (unsigned)(l * SAMPK + j) * 0x9E3779B9u + 0x85EBCA6Bu;
    x ^= x >> 16; x *= 0x7feb352du; x ^= x >> 15; x *= 0x846ca68bu; x ^= x >> 16;
    int idx = (int)(x & (S - 1));
    const float* krow = K + ((size_t)((b * L + idx) * H + h)) * D;
    float acc = 0.f;
#pragma unroll
    for (int t = 0; t < 16; ++t) {
      float4 kv = ((const float4*)krow)[t];
      acc += q[t].x * kv.x + q[t].y * kv.y + q[t].z * kv.z + q[t].w * kv.w;
    }
    mx = fmaxf(mx, acc);
    sm += acc;
  }
  M[(size_t)bh * L + l] = mx - sm * (1.0f / (float)S);
}

// =====================================================================
// Kernel 2: top-38 per (b,h): iterative argmax (stable tie-break = low idx)
// =====================================================================
__global__ __launch_bounds__(256) void k_topk(const float* __restrict__ M,
                                              int* __restrict__ Mtop) {
  __shared__ float sv[L];
  __shared__ float rv[256];
  __shared__ int ri[256];
  int bh = blockIdx.x;
  int tid = threadIdx.x;
  for (int k = tid; k < L; k += 256) sv[k] = M[(size_t)bh * L + k];
  __syncthreads();
  for (int it = 0; it < NTOP; ++it) {
    float best = -__builtin_inff();
    int bi = 0;
    for (int k = tid; k < L; k += 256) {
      float v = sv[k];
      if (v > best) { best = v; bi = k; }
    }
    rv[tid] = best; ri[tid] = bi;
    __syncthreads();
    for (int off = 128; off > 0; off >>= 1) {
      if (tid < off) {
        float v2 = rv[tid + off]; int i2 = ri[tid + off];
        if (v2 > rv[tid] || (v2 == rv[tid] && i2 < ri[tid])) { rv[tid] = v2; ri[tid] = i2; }
      }
      __syncthreads();
    }
    if (tid == 0) {
      Mtop[bh * NTOP_PAD + it] = ri[0];
      sv[ri[0]] = -__builtin_inff();
    }
    __syncthreads();
  }
  if (tid == 0)
    for (int it = NTOP; it < NTOP_PAD; ++it) Mtop[bh * NTOP_PAD + it] = -1;
}

// =====================================================================
// Kernels 3a/3b/3c: blocked cumsum of V along L  (the bandwidth-bound part)
// =====================================================================
__global__ __launch_bounds__(64) void k_partial(const float* __restrict__ V,
                                                float* __restrict__ part) {
  int c = blockIdx.x % NCHUNK;
  int bh = blockIdx.x / NCHUNK;
  int h = bh % H, b = bh / H;
  int d = threadIdx.x;
  const float* base = V + ((size_t)((b * L + c * CHUNK) * H + h)) * D + d;
  float s = 0.f;
#pragma unroll 8
  for (int r = 0; r < CHUNK; ++r) s += base[(size_t)r * H * D];
  part[((size_t)bh * NCHUNK + c) * D + d] = s;
}

__global__ __launch_bounds__(64) void k_scanpart(float* __restrict__ part) {
  int bh = blockIdx.x;
  int d = threadIdx.x;
  float run = 0.f;
  for (int c = 0; c < NCHUNK; ++c) {
    size_t i = ((size_t)bh * NCHUNK + c) * D + d;
    float v = part[i];
    part[i] = run;
    run += v;
  }
}

__global__ __launch_bounds__(64) void k_cumsum(const float* __restrict__ V,
                                               const float* __restrict__ part,
                                               float* __restrict__ out) {
  int c = blockIdx.x % NCHUNK;
  int bh = blockIdx.x / NCHUNK;
  int h = bh % H, b = bh / H;
  int d = threadIdx.x;
  float run = part[((size_t)bh * NCHUNK + c) * D + d];
  const float* vin = V + ((size_t)((b * L + c * CHUNK) * H + h)) * D + d;
  float* vout = out + ((size_t)((b * L + c * CHUNK) * H + h)) * D + d;
  for (int g = 0; g < CHUNK / 8; ++g) {
    float v[8];
#pragma unroll
    for (int i = 0; i < 8; ++i) v[i] = vin[(size_t)(g * 8 + i) * H * D];
#pragma unroll
    for (int i = 0; i < 8; ++i) {
      run += v[i];
      vout[(size_t)(g * 8 + i) * H * D] = run;
    }
  }
}

// =====================================================================
// Kernel 4: flash attention for the 38 selected queries, f32 WMMA 16x16x4.
// One wave32 per (b, h, 16-query tile). Streams 128 key tiles of 16.
// =====================================================================
__global__ __launch_bounds__(32) void k_sparse_attn(const float* __restrict__ Q,
                                                    const float* __restrict__ K,
                                                    const float* __restrict__ V,
                                                    const int* __restrict__ Mtop,
                                                    float* __restrict__ out) {
  constexpr int QT = 3;
  int qt = blockIdx.x % QT;
  int bh = blockIdx.x / QT;
  int h = bh % H, b = bh / H;
  int lane = (int)threadIdx.x;
  int half = lane >> 4;   // 0: K pair {0,1} / rows 0..7 ; 1: K pair {2,3} / rows 8..15
  int n = lane & 15;      // A-frag row / B-,C-frag column
  const size_t rowstride = (size_t)H * D;

  const v2f ZERO2 = {};

  // ---- A-fragment (Q) for query row n of this tile; per-lane pair Q[row][4t+2*half .. +1]
  int ua = qt * 16 + n;
  int qpos = (ua < NTOP) ? Mtop[bh * NTOP_PAD + ua] : -1;
  v2f qf[16];
  if (qpos >= 0) {
    const float* qp = Q + ((size_t)(b * L + qpos) * H + h) * D;
#pragma unroll
    for (int t = 0; t < 16; ++t) qf[t] = *(const v2f*)(qp + 4 * t + 2 * half);
  } else {
#pragma unroll
    for (int t = 0; t < 16; ++t) qf[t] = ZERO2;
  }

  // ---- causal mask positions for the 8 C-layout rows (M = 8*half + r) this lane owns
  int mpos[8];
#pragma unroll
  for (int r = 0; r < 8; ++r) {
    int uu = qt * 16 + 8 * half + r;
    mpos[r] = (uu < NTOP) ? Mtop[bh * NTOP_PAD + uu] : (1 << 30);  // padded rows: unmasked
  }

  float mval[8], ssum[8];
#pragma unroll
  for (int r = 0; r < 8; ++r) { mval[r] = -__builtin_inff(); ssum[r] = 0.f; }
  v8f o[4] = {};  // O[16 x 64] as four 16x16 C tiles

  __shared__ __align__(16) float pl[16 * 16];  // P tile D-layout -> A-layout shuttle

  const float* Kbh = K + ((size_t)(b * L) * H + h) * D;
  const float* Vbh = V + ((size_t)(b * L) * H + h) * D;

  for (int kb = 0; kb < S / 16; ++kb) {
    // ---- B-fragments (K^T): per-lane pair K[key = kb*16+n][4t+2*half .. +1]
    const float* krow = Kbh + (size_t)(kb * 16 + n) * rowstride;
    v2f kf[16];
#pragma unroll
    for (int t = 0; t < 16; ++t) kf[t] = *(const v2f*)(krow + 4 * t + 2 * half);

    // ---- scores tile: 16 WMMAs over K dim (64 = 16 * 4)
    v8f c = {};
#pragma unroll
    for (int t = 0; t < 16; ++t) c = wmma_f32_16x16x4(qf[t], kf[t], c);

    // ---- scale + causal mask + online softmax (row = 8*half + r, col = kb*16 + n)
    int col = kb * 16 + n;
    float p8[8], alpha8[8];
#pragma unroll
    for (int r = 0; r < 8; ++r) {
      float sc = c[r] * SCALE;
      if (col > mpos[r]) sc = -__builtin_inff();
      float rm = sc;  // row max across the 16 lanes of this half
      rm = fmaxf(rm, __shfl_xor(rm, 1));
      rm = fmaxf(rm, __shfl_xor(rm, 2));
      rm = fmaxf(rm, __shfl_xor(rm, 4));
      rm = fmaxf(rm, __shfl_xor(rm, 8));
      float nm = fmaxf(mval[r], rm);
      float alpha, p;
      if (nm == -__builtin_inff()) { alpha = 1.f; p = 0.f; }
      else { alpha = __expf(mval[r] - nm); p = __expf(sc - nm); }
      mval[r] = nm;
      float ps = p;  // row sum
      ps += __shfl_xor(ps, 1);
      ps += __shfl_xor(ps, 2);
      ps += __shfl_xor(ps, 4);
      ps += __shfl_xor(ps, 8);
      ssum[r] = ssum[r] * alpha + ps;
      p8[r] = p;
      alpha8[r] = alpha;
    }
#pragma unroll
    for (int dt = 0; dt < 4; ++dt)
#pragma unroll
      for (int r = 0; r < 8; ++r) o[dt][r] *= alpha8[r];

    // ---- D-layout -> A-layout through LDS
    __syncthreads();  // WAR vs previous iteration's reads
#pragma unroll
    for (int r = 0; r < 8; ++r) pl[(8 * half + r) * 16 + n] = p8[r];
    __syncthreads();

    v2f pa[4];  // P A-fragments: P[row=n][4*ks + 2*half .. +1]
#pragma unroll
    for (int ks = 0; ks < 4; ++ks) pa[ks] = *(const v2f*)(&pl[n * 16 + 4 * ks + 2 * half]);

    // ---- O += P(16x16) @ V(16x64): 4 D-tiles x 4 K-steps
#pragma unroll
    for (int dt = 0; dt < 4; ++dt) {
#pragma unroll
      for (int ks = 0; ks < 4; ++ks) {
        const float* vp = Vbh + (size_t)(kb * 16 + 4 * ks + 2 * half) * rowstride + 16 * dt + n;
        v2f bv;
        bv.x = vp[0];
        bv.y = vp[rowstride];
        o[dt] = wmma_f32_16x16x4(pa[ks], bv, o[dt]);
      }
    }
  }

  // ---- normalize + scatter into the cumsum output at the selected rows
#pragma unroll
  for (int r = 0; r < 8; ++r) {
    int uu = qt * 16 + 8 * half + r;
    if (uu < NTOP) {
      float inv = 1.f / ssum[r];
      float* op = out + ((size_t)(b * L + mpos[r]) * H + h) * D + n;
#pragma unroll
      for (int dt = 0; dt < 4; ++dt) op[16 * dt] = o[dt][r] * inv;
    }
  }
}

// =====================================================================
// Host launcher
// =====================================================================
extern "C" void kernel_launch(void* const* d_in, const int* in_sizes, int n_in,
                              void* d_out, int out_size, void* d_ws, size_t ws_size,
                              hipStream_t stream) {
  const float* Q = (const float*)d_in[0];
  const float* K = (const float*)d_in[1];
  const float* V = (const float*)d_in[2];
  float* out = (float*)d_out;

  float* Mws = (float*)d_ws;                                // B*H*L floats
  int* Mtop = (int*)(Mws + (size_t)B * H * L);              // B*H*NTOP_PAD ints
  float* part = (float*)(Mtop + (size_t)B * H * NTOP_PAD);  // B*H*NCHUNK*D floats

  k_sample_m<<<(B * H * L) / 256, 256, 0, stream>>>(Q, K, Mws);
  k_topk<<<B * H, 256, 0, stream>>>(Mws, Mtop);
  k_partial<<<B * H * NCHUNK, 64, 0, stream>>>(V, part);
  k_scanpart<<<B * H, 64, 0, stream>>>(part);
  k_cumsum<<<B * H * NCHUNK, 64, 0, stream>>>(V, part, out);
  k_sparse_attn<<<B * H * 3, 32, 0, stream>>>(Q, K, V, Mtop, out);
}